// HeteAttention_53626961658091
// MI455X (gfx1250) — compile-verified
//
#include <hip/hip_runtime.h>
#include <math.h>

// Problem constants (match reference)
#define NN   50000
#define EE   400000
#define RR   3
#define INF_ 128
#define OUTF 128
#define HH   4
#define DD   32
#define SLOPE 0.2f

#define MTILES      3125            // NN / 16 exactly
#define TPW         8               // M tiles per wave
#define WAVES_PB    4               // waves per block
#define TILES_PB    (TPW * WAVES_PB)

typedef float v2f __attribute__((ext_vector_type(2)));
typedef float v8f __attribute__((ext_vector_type(8)));

// ---------------------------------------------------------------------------
// Utility: fill buffer with a constant
// ---------------------------------------------------------------------------
__global__ void fill_kernel(float* __restrict__ p, size_t n, float v) {
    size_t i = (size_t)blockIdx.x * blockDim.x + threadIdx.x;
    if (i < n) p[i] = v;
}

// ---------------------------------------------------------------------------
// float atomic max via int bit trick (m initialized to -inf)
// ---------------------------------------------------------------------------
__device__ __forceinline__ void atomicMaxFloat(float* addr, float val) {
    if (val >= 0.0f) {
        atomicMax((int*)addr, __float_as_int(val));
    } else {
        atomicMin((unsigned int*)addr, __float_as_uint(val));
    }
}

// ---------------------------------------------------------------------------
// Kernel 1: fs/fd projections via fp32 WMMA  (D = X @ W + b)
//
//   grid.y = 6 (relation*2 + src/dst), grid.x = ceil(3125 / 32)
//   block  = 128 threads = 4 waves; each wave owns 8 sequential 16-row
//   M tiles (full 16x128 strips: 8 N-tiles x 32 K-steps of
//   v_wmma_f32_16x16x4_f32 each).
//
//   W (128x128) is staged ONCE per block into LDS, pre-swizzled into the
//   B-fragment pair layout:  (k,n) -> ldsW[(k&~1)*128 + 2n + (k&1)]
//   so each B fragment is a single 8B-aligned ds_load_b64.
//   A fragments are 8B v2f loads straight from global (X is L2-resident).
// ---------------------------------------------------------------------------
__global__ __launch_bounds__(128)
void gemm_proj_kernel(const float* __restrict__ x,
                      const float* __restrict__ W_src, const float* __restrict__ b_src,
                      const float* __restrict__ W_dst, const float* __restrict__ b_dst,
                      float* __restrict__ fs, float* __restrict__ fd)
{
    const int wave = threadIdx.x >> 5;
    const int lane = threadIdx.x & 31;
    const int half = lane >> 4;     // 0: lanes 0-15, 1: lanes 16-31
    const int lm   = lane & 15;

    const int sel   = blockIdx.y;   // 0..5
    const int r     = sel >> 1;
    const bool isD  = sel & 1;
    const float* W    = (isD ? W_dst : W_src) + (size_t)r * INF_ * OUTF;
    const float* bias = (isD ? b_dst : b_src) + r * OUTF;
    float* out        = (isD ? fd : fs) + (size_t)r * NN * OUTF;

    // ---- stage W into LDS, packed for B fragments (64 KB) ----
    __shared__ float ldsW[INF_ * OUTF];
    for (int base = threadIdx.x * 4; base < INF_ * OUTF; base += 128 * 4) {
        const int k = base >> 7;        // row
        const int n = base & 127;       // col (multiple of 4)
        float4 v = *(const float4*)(W + base);
        float* dp = ldsW + (size_t)(k & ~1) * OUTF + 2 * n + (k & 1);
        dp[0] = v.x; dp[2] = v.y; dp[4] = v.z; dp[6] = v.w;
    }
    __syncthreads();

    const float bv[1] = { 0.0f }; (void)bv;

    const int tile0 = blockIdx.x * TILES_PB + wave * TPW;

    for (int ti = 0; ti < TPW; ++ti) {
        const int tile = tile0 + ti;
        if (tile >= MTILES) break;
        const int m0 = tile * 16;

        // prefetch next tile's X rows (8 KB = 128 cachelines, 4 per lane)
        if (ti + 1 < TPW && tile + 1 < MTILES) {
            const float* nx = x + (size_t)(tile + 1) * 16 * INF_;
            #pragma unroll
            for (int i = 0; i < 4; ++i)
                __builtin_prefetch(nx + (size_t)(lane + 32 * i) * 16, 0, 3);
        }

        v8f acc[8];
        #pragma unroll
        for (int t = 0; t < 8; ++t) acc[t] = (v8f){0,0,0,0,0,0,0,0};

        const float* arow = x + ((size_t)m0 + lm) * INF_;

        for (int k = 0; k < INF_; k += 4) {
            const int kk = k + 2 * half;            // even
            // A fragment: row lm, cols kk,kk+1 -> one 8B global load
            v2f a = *(const v2f*)(arow + kk);
            const float* bbase = ldsW + (size_t)kk * OUTF + 2 * lm;
            #pragma unroll
            for (int t = 0; t < 8; ++t) {
                v2f b = *(const v2f*)(bbase + 32 * t);
                acc[t] = __builtin_amdgcn_wmma_f32_16x16x4_f32(
                    false, a, false, b, (short)0, acc[t], false, false);
            }
        }

        // Store + bias. C/D layout: VGPR j -> row m0 + j + 8*half, col t*16+lm
        #pragma unroll
        for (int t = 0; t < 8; ++t) {
            const int col = t * 16 + lm;
            const float b0 = bias[col];
            #pragma unroll
            for (int j = 0; j < 8; ++j) {
                const int row = m0 + j + 8 * half;
                out[(size_t)row * OUTF + col] = acc[t][j] + b0;
            }
        }
    }
}

// ---------------------------------------------------------------------------
// Kernel 2: per-(r,e,h) attention logit s = a_h . leakyrelu(fs[src]+fd[dst])
//           and segment max into m[r][dst][h]   (float4-vectorized gathers)
// ---------------------------------------------------------------------------
__global__ void edge_logits_kernel(const int* __restrict__ src, const int* __restrict__ dst,
                                   const float* __restrict__ fs, const float* __restrict__ fd,
                                   const float* __restrict__ attn,
                                   float* __restrict__ sbuf, float* __restrict__ mbuf)
{
    size_t idx = (size_t)blockIdx.x * blockDim.x + threadIdx.x;
    const size_t total = (size_t)RR * EE * HH;
    if (idx >= total) return;
    const int h = (int)(idx % HH);
    size_t t = idx / HH;
    const int e = (int)(t % EE);
    const int r = (int)(t / EE);

    const int sn = src[(size_t)r * EE + e];
    const int dn = dst[(size_t)r * EE + e];
    const float4* pfs = (const float4*)(fs + ((size_t)r * NN + sn) * OUTF + h * DD);
    const float4* pfd = (const float4*)(fd + ((size_t)r * NN + dn) * OUTF + h * DD);
    const float4* pa  = (const float4*)(attn + (size_t)(r * HH + h) * DD);

    float acc = 0.0f;
    #pragma unroll
    for (int q = 0; q < DD / 4; ++q) {
        const float4 a_ = pfs[q];
        const float4 b_ = pfd[q];
        const float4 w_ = pa[q];
        float v;
        v = a_.x + b_.x; v = (v > 0.0f) ? v : SLOPE * v; acc = fmaf(v, w_.x, acc);
        v = a_.y + b_.y; v = (v > 0.0f) ? v : SLOPE * v; acc = fmaf(v, w_.y, acc);
        v = a_.z + b_.z; v = (v > 0.0f) ? v : SLOPE * v; acc = fmaf(v, w_.z, acc);
        v = a_.w + b_.w; v = (v > 0.0f) ? v : SLOPE * v; acc = fmaf(v, w_.w, acc);
    }
    sbuf[idx] = acc;
    atomicMaxFloat(&mbuf[((size_t)r * NN + dn) * HH + h], acc);
}

// ---------------------------------------------------------------------------
// Kernel 3: p = exp(s - m[dst]); segment sum into z[r][dst][h]
// ---------------------------------------------------------------------------
__global__ void edge_exp_kernel(const int* __restrict__ dst,
                                const float* __restrict__ sbuf, const float* __restrict__ mbuf,
                                float* __restrict__ pbuf, float* __restrict__ zbuf)
{
    size_t idx = (size_t)blockIdx.x * blockDim.x + threadIdx.x;
    const size_t total = (size_t)RR * EE * HH;
    if (idx >= total) return;
    const int h = (int)(idx % HH);
    size_t t = idx / HH;
    const int e = (int)(t % EE);
    const int r = (int)(t / EE);

    const int dn = dst[(size_t)r * EE + e];
    const float m = mbuf[((size_t)r * NN + dn) * HH + h];   // finite: dn has edges
    const float p = __expf(sbuf[idx] - m);
    pbuf[idx] = p;
    atomicAdd(&zbuf[((size_t)r * NN + dn) * HH + h], p);
}

// ---------------------------------------------------------------------------
// Kernel 4: out[r][dst][o] += (p/z) * fs[src][o]   (one thread per (r,e,o))
// ---------------------------------------------------------------------------
__global__ void edge_agg_kernel(const int* __restrict__ src, const int* __restrict__ dst,
                                const float* __restrict__ fs,
                                const float* __restrict__ pbuf, const float* __restrict__ zbuf,
                                float* __restrict__ hout)
{
    size_t idx = (size_t)blockIdx.x * blockDim.x + threadIdx.x;
    const size_t total = (size_t)RR * EE * OUTF;
    if (idx >= total) return;
    const int o = (int)(idx % OUTF);
    size_t t = idx / OUTF;
    const int e = (int)(t % EE);
    const int r = (int)(t / EE);
    const int h = o >> 5;           // DD = 32

    const int sn = src[(size_t)r * EE + e];
    const int dn = dst[(size_t)r * EE + e];
    const float p = pbuf[((size_t)r * EE + e) * HH + h];
    const float z = zbuf[((size_t)r * NN + dn) * HH + h];
    const float alpha = p / fmaxf(z, 1e-9f);
    const float v = alpha * fs[((size_t)r * NN + sn) * OUTF + o];
    atomicAdd(&hout[((size_t)r * NN + dn) * OUTF + o], v);
}

// ---------------------------------------------------------------------------
// Kernel 5: relation attention scores, fused:
//   wsum[r] += sum_n ( tanh(h[r][n] @ P1 + p1b) . P2 )
//   8 rows per block; P1 loaded once per k per block.
// ---------------------------------------------------------------------------
__global__ __launch_bounds__(128)
void relproj_kernel(const float* __restrict__ hout, const float* __restrict__ P1,
                    const float* __restrict__ p1b, const float* __restrict__ P2,
                    float* __restrict__ wsum)
{
    const int o = threadIdx.x;                  // 0..127
    const int blocksPerRel = NN / 8;            // 6250
    const int r  = blockIdx.x / blocksPerRel;
    const int n0 = (blockIdx.x % blocksPerRel) * 8;
    const float* base = hout + ((size_t)r * NN + n0) * OUTF;

    __shared__ float sh[8][OUTF];
    for (int j = 0; j < 8; ++j) sh[j][o] = base[(size_t)j * OUTF + o];
    __syncthreads();

    float acc[8];
    const float bo = p1b[o];
    #pragma unroll
    for (int j = 0; j < 8; ++j) acc[j] = bo;
    for (int k = 0; k < OUTF; ++k) {
        const float p = P1[(size_t)k * OUTF + o];
        #pragma unroll
        for (int j = 0; j < 8; ++j) acc[j] = fmaf(sh[j][k], p, acc[j]);
    }
    const float p2 = P2[o];
    float tot = 0.0f;
    #pragma unroll
    for (int j = 0; j < 8; ++j) tot += tanhf(acc[j]) * p2;

    __shared__ float red[128];
    red[o] = tot;
    __syncthreads();
    for (int st = 64; st > 0; st >>= 1) {
        if (o < st) red[o] += red[o + st];
        __syncthreads();
    }
    if (o == 0) atomicAdd(&wsum[r], red[0]);
}

// ---------------------------------------------------------------------------
// Kernel 6: beta = softmax(wsum / N)
// ---------------------------------------------------------------------------
__global__ void beta_kernel(const float* __restrict__ wsum, float* __restrict__ beta) {
    if (threadIdx.x == 0 && blockIdx.x == 0) {
        float w0 = wsum[0] / (float)NN;
        float w1 = wsum[1] / (float)NN;
        float w2 = wsum[2] / (float)NN;
        float mx = fmaxf(w0, fmaxf(w1, w2));
        float e0 = __expf(w0 - mx), e1 = __expf(w1 - mx), e2 = __expf(w2 - mx);
        float s = e0 + e1 + e2;
        beta[0] = e0 / s; beta[1] = e1 / s; beta[2] = e2 / s;
    }
}

// ---------------------------------------------------------------------------
// Kernel 7: out = relu(sum_r beta_r * h[r]) + h_bias
// ---------------------------------------------------------------------------
__global__ void final_kernel(const float* __restrict__ hout, const float* __restrict__ beta,
                             const float* __restrict__ h_bias, float* __restrict__ out)
{
    size_t idx = (size_t)blockIdx.x * blockDim.x + threadIdx.x;
    const size_t total = (size_t)NN * OUTF;
    if (idx >= total) return;
    const float b0 = beta[0], b1 = beta[1], b2 = beta[2];
    float v = b0 * hout[idx]
            + b1 * hout[(size_t)NN * OUTF + idx]
            + b2 * hout[2 * (size_t)NN * OUTF + idx];
    out[idx] = fmaxf(v, 0.0f) + h_bias[idx & (OUTF - 1)];
}

// ---------------------------------------------------------------------------
extern "C" void kernel_launch(void* const* d_in, const int* in_sizes, int n_in,
                              void* d_out, int out_size, void* d_ws, size_t ws_size,
                              hipStream_t stream) {
    const float* x      = (const float*)d_in[0];
    const int*   src    = (const int*)  d_in[1];
    const int*   dst    = (const int*)  d_in[2];
    const float* W_src  = (const float*)d_in[3];
    const float* b_src  = (const float*)d_in[4];
    const float* W_dst  = (const float*)d_in[5];
    const float* b_dst  = (const float*)d_in[6];
    const float* attn   = (const float*)d_in[7];
    const float* P1     = (const float*)d_in[8];
    const float* p1b    = (const float*)d_in[9];
    const float* P2     = (const float*)d_in[10];
    const float* h_bias = (const float*)d_in[11];
    float* out = (float*)d_out;

    // Workspace layout (floats). hout aliases fd (fd dead after logit pass).
    float* ws = (float*)d_ws;
    size_t off = 0;
    float* fs   = ws + off; off += (size_t)RR * NN * OUTF;   // 19.2M
    float* fd   = ws + off; off += (size_t)RR * NN * OUTF;   // 19.2M
    float* hout = fd;                                        // alias
    float* sbuf = ws + off; off += (size_t)RR * EE * HH;     // 4.8M
    float* pbuf = ws + off; off += (size_t)RR * EE * HH;     // 4.8M
    float* mbuf = ws + off; off += (size_t)RR * NN * HH;     // 0.6M
    float* zbuf = ws + off; off += (size_t)RR * NN * HH;     // 0.6M
    float* wsum = ws + off; off += 4;
    float* beta = ws + off; off += 4;

    const int TB = 256;
    const size_t nmh = (size_t)RR * NN * HH;

    // init m = -inf, z = 0, wsum = 0
    fill_kernel<<<(unsigned)((nmh + TB - 1) / TB), TB, 0, stream>>>(mbuf, nmh, -INFINITY);
    fill_kernel<<<(unsigned)((nmh + TB - 1) / TB), TB, 0, stream>>>(zbuf, nmh, 0.0f);
    fill_kernel<<<1, 32, 0, stream>>>(wsum, 4, 0.0f);

    // 1) projections (WMMA): 3125 M-tiles, 32 tiles/block, 6 GEMM selections
    {
        dim3 grid((MTILES + TILES_PB - 1) / TILES_PB, 6);
        gemm_proj_kernel<<<grid, 128, 0, stream>>>(x, W_src, b_src, W_dst, b_dst, fs, fd);
    }

    // 2) logits + segment max
    {
        const size_t total = (size_t)RR * EE * HH;
        edge_logits_kernel<<<(unsigned)((total + TB - 1) / TB), TB, 0, stream>>>(
            src, dst, fs, fd, attn, sbuf, mbuf);
    }

    // 3) exp + segment sum
    {
        const size_t total = (size_t)RR * EE * HH;
        edge_exp_kernel<<<(unsigned)((total + TB - 1) / TB), TB, 0, stream>>>(
            dst, sbuf, mbuf, pbuf, zbuf);
    }

    // zero hout (aliases fd, which is no longer needed)
    {
        const size_t total = (size_t)RR * NN * OUTF;
        fill_kernel<<<(unsigned)((total + TB - 1) / TB), TB, 0, stream>>>(hout, total, 0.0f);
    }

    // 4) alpha-weighted scatter aggregation
    {
        const size_t total = (size_t)RR * EE * OUTF;
        edge_agg_kernel<<<(unsigned)((total + TB - 1) / TB), TB, 0, stream>>>(
            src, dst, fs, pbuf, zbuf, hout);
    }

    // 5) relation score sums
    relproj_kernel<<<RR * (NN / 8), 128, 0, stream>>>(hout, P1, p1b, P2, wsum);

    // 6) beta softmax
    beta_kernel<<<1, 32, 0, stream>>>(wsum, beta);

    // 7) final blend + relu + bias
    {
        const size_t total = (size_t)NN * OUTF;
        final_kernel<<<(unsigned)((total + TB - 1) / TB), TB, 0, stream>>>(
            hout, beta, h_bias, out);
    }
}